// ReverbNetwork_43173011259521
// MI455X (gfx1250) — compile-verified
//
#include <hip/hip_runtime.h>

typedef __attribute__((ext_vector_type(2))) float v2f;
typedef __attribute__((ext_vector_type(8))) float v8f;

#define HW     256
#define CH     3
#define BATCH  16
#define KS     4
#define CKK    48                       // CH*KS*KS
#define NEDGE  11
#define NNODE  8
#define CSTR   (HW*HW)                  // 65536
#define BSTR   (CH*CSTR)                // 196608
#define NSTR   (BATCH*BSTR)             // 3145728
#define NEWSTATES_ELEMS (NNODE*NSTR)    // 25165824
#define WCNT   144                      // 3*3*4*4 per edge
#define ALPHA  0.1f

__constant__ int   c_src[NEDGE]          = {0,1,2,3,4,5,6,7,0,2,8};
__constant__ int   c_dst[NEDGE]          = {1,2,3,4,5,6,7,0,4,6,0};
__constant__ int   c_edge_cnt[NNODE]     = {2,1,1,1,2,1,2,1};
__constant__ int   c_edge_list[NNODE][2] = {{7,10},{0,0},{1,0},{2,0},{3,8},{4,0},{5,9},{6,0}};
__constant__ float c_invindeg[NNODE]     = {0.5f,1.f,1.f,1.f,0.5f,1.f,0.5f,1.f};

__device__ __forceinline__ float sigf(float x) {
    return 1.0f / (1.0f + __expf(-x));
}

// ---------------------------------------------------------------------------
// Kernel 0: scaled effective weights  wsW[e][o][c] = (w + a*p) / indeg(dst[e])
//           and zero the hebb accumulator.
// ---------------------------------------------------------------------------
__global__ void prep_kernel(const float* __restrict__ weights,
                            const float* __restrict__ plastic,
                            float* __restrict__ wsW,
                            float* __restrict__ wsH) {
    int t = blockIdx.x * blockDim.x + threadIdx.x;
    if (t < NEDGE * WCNT) {
        int e = t / WCNT;
        wsW[t] = (weights[t] + ALPHA * plastic[t]) * c_invindeg[c_dst[e]];
        wsH[t] = 0.f;
    }
}

// ---------------------------------------------------------------------------
// Kernel 1: per-row implicit-GEMM conv + dest aggregation.
// Workgroup = (node n, batch b, row h); 8 waves, 32 pixels / wave.
// D[m=pixel16, n=outch] += A[m, k=(i,kh,kw)] * B[k, n]  via 12x WMMA f32 16x16x4
// ---------------------------------------------------------------------------
__launch_bounds__(256)
__global__ void conv_agg_kernel(const float* __restrict__ states,
                                const float* __restrict__ wsW,
                                float* __restrict__ outStates) {
    __shared__ float sPre[CH][KS][264];   // rows h-1..h+2, cols -1..258 (zero-padded)
    __shared__ float sOut[CH][HW];

    const int h = blockIdx.x, b = blockIdx.y, n = blockIdx.z;
    const int tid   = threadIdx.x;
    const int lane  = tid & 31;
    const int wave  = tid >> 5;
    const int nIdx  = lane & 15;     // m for A, n for B/D
    const int khalf = lane >> 4;     // selects k=2*khalf+r within a K=4 step

    v8f acc[2] = {};                 // two 16-pixel M-tiles per wave
    const int ecnt = c_edge_cnt[n];

    for (int ei = 0; ei < ecnt; ++ei) {
        const int e = c_edge_list[n][ei];
        const int s = c_src[e];
        if (ei) __syncthreads();     // protect sPre reuse
        // stage sigmoid(states[src]) strip: 3 ch x 4 rows x 260 cols
        for (int idx = tid; idx < CH * KS * 260; idx += 256) {
            int i   = idx / (KS * 260);
            int rem = idx - i * (KS * 260);
            int rr  = rem / 260;
            int cc  = rem - rr * 260;
            int hh  = h - 1 + rr;
            int ww  = cc - 1;
            float v = 0.f;
            if ((unsigned)hh < HW && (unsigned)ww < HW)
                v = sigf(states[s * NSTR + b * BSTR + i * CSTR + hh * HW + ww]);
            sPre[i][rr][cc] = v;
        }
        __syncthreads();

        // B fragments: W[c, o] for the 12 K-steps (constant per edge)
        v2f Bf[12];
        #pragma unroll
        for (int s12 = 0; s12 < 12; ++s12) {
            #pragma unroll
            for (int r = 0; r < 2; ++r) {
                int k = 2 * khalf + r;
                int c = 4 * s12 + k;
                Bf[s12][r] = (nIdx < CH) ? wsW[e * WCNT + nIdx * CKK + c] : 0.f;
            }
        }

        #pragma unroll
        for (int t = 0; t < 2; ++t) {
            const int wbase = wave * 32 + t * 16;
            #pragma unroll
            for (int s12 = 0; s12 < 12; ++s12) {
                v2f Af;
                #pragma unroll
                for (int r = 0; r < 2; ++r) {
                    int k  = 2 * khalf + r;
                    int c  = 4 * s12 + k;          // c = i*16 + kh*4 + kw
                    int i  = c >> 4;
                    int kh = (c >> 2) & 3;
                    int kw = c & 3;
                    Af[r] = sPre[i][kh][wbase + nIdx + kw];
                }
                acc[t] = __builtin_amdgcn_wmma_f32_16x16x4_f32(
                    false, Af, false, Bf[s12], (short)0, acc[t], false, false);
            }
        }
    }

    // scatter D (lanes n<3 hold all data) -> LDS, then coalesced store
    #pragma unroll
    for (int t = 0; t < 2; ++t) {
        if (nIdx < CH) {
            #pragma unroll
            for (int v = 0; v < 8; ++v) {
                int m = v + 8 * khalf;
                sOut[nIdx][wave * 32 + t * 16 + m] = acc[t][v];
            }
        }
    }
    __syncthreads();
    for (int idx = tid; idx < CH * HW; idx += 256) {
        int o = idx >> 8, w = idx & 255;
        outStates[n * NSTR + b * BSTR + o * CSTR + h * HW + w] = sOut[o][w];
    }
}

// ---------------------------------------------------------------------------
// Kernel 2: Hebbian correlation = GEMM with K = pixels.
// Workgroup = (edge e, batch b, 8-row block); each wave owns one row.
// D[m=c(48: 3 full M-tiles), n=o] += A[m, k=4 px] * B[k=4 px, n]
// ---------------------------------------------------------------------------
__launch_bounds__(256)
__global__ void hebb_kernel(const float* __restrict__ states,
                            const float* __restrict__ newStates,
                            float* __restrict__ wsH) {
    __shared__ float sPre[CH][11][264];   // pre rows h0-1..h0+9, cols -1..258
    __shared__ float sPost[CH][8][HW];    // post rows h0..h0+7
    __shared__ float sHebb[WCNT];

    const int h0 = blockIdx.x * 8;
    const int b  = blockIdx.y;
    const int e  = blockIdx.z;
    const int s  = c_src[e];
    const int d  = c_dst[e];
    const int tid   = threadIdx.x;
    const int lane  = tid & 31;
    const int wave  = tid >> 5;
    const int nIdx  = lane & 15;
    const int khalf = lane >> 4;

    for (int idx = tid; idx < CH * 11 * 260; idx += 256) {
        int i   = idx / (11 * 260);
        int rem = idx - i * (11 * 260);
        int rr  = rem / 260;
        int cc  = rem - rr * 260;
        int hh  = h0 - 1 + rr;
        int ww  = cc - 1;
        float v = 0.f;
        if ((unsigned)hh < HW && (unsigned)ww < HW)
            v = sigf(states[s * NSTR + b * BSTR + i * CSTR + hh * HW + ww]);
        sPre[i][rr][cc] = v;
    }
    for (int idx = tid; idx < CH * 8 * HW; idx += 256) {
        int i   = idx / (8 * HW);
        int rem = idx - i * (8 * HW);
        int rr  = rem >> 8, ww = rem & 255;
        sPost[i][rr][ww] =
            sigf(newStates[d * NSTR + b * BSTR + i * CSTR + (h0 + rr) * HW + ww]);
    }
    if (tid < WCNT) sHebb[tid] = 0.f;
    __syncthreads();

    v8f acc[CH] = {};                 // 3 M-tiles covering c = 0..47
    for (int w0 = 0; w0 < HW; w0 += 4) {
        v2f Bf;                       // post[o] at 4 consecutive pixels
        #pragma unroll
        for (int r = 0; r < 2; ++r) {
            int k = 2 * khalf + r;
            Bf[r] = (nIdx < CH) ? sPost[nIdx][wave][w0 + k] : 0.f;
        }
        #pragma unroll
        for (int t = 0; t < CH; ++t) {   // i == t; c = t*16 + nIdx
            v2f Af;
            int kh = nIdx >> 2, kw = nIdx & 3;
            #pragma unroll
            for (int r = 0; r < 2; ++r) {
                int k = 2 * khalf + r;
                Af[r] = sPre[t][wave + kh][w0 + k + kw];
            }
            acc[t] = __builtin_amdgcn_wmma_f32_16x16x4_f32(
                false, Af, false, Bf, (short)0, acc[t], false, false);
        }
    }

    // 8-wave LDS reduction, then one global atomic pass per workgroup
    #pragma unroll
    for (int t = 0; t < CH; ++t) {
        if (nIdx < CH) {
            #pragma unroll
            for (int v = 0; v < 8; ++v) {
                int m = v + 8 * khalf;                 // c = t*16 + m
                atomicAdd(&sHebb[nIdx * CKK + t * 16 + m], acc[t][v]);
            }
        }
    }
    __syncthreads();
    if (tid < WCNT) atomicAdd(&wsH[e * WCNT + tid], sHebb[tid]);
}

// ---------------------------------------------------------------------------
// Kernel 3: new_plastic = plastic + alpha * hebb_sum / (B*H*W)
// ---------------------------------------------------------------------------
__global__ void finish_kernel(const float* __restrict__ plastic,
                              const float* __restrict__ wsH,
                              float* __restrict__ outPlastic) {
    int t = blockIdx.x * blockDim.x + threadIdx.x;
    if (t < NEDGE * WCNT)
        outPlastic[t] = plastic[t] + (ALPHA / (float)(BATCH * CSTR)) * wsH[t];
}

extern "C" void kernel_launch(void* const* d_in, const int* in_sizes, int n_in,
                              void* d_out, int out_size, void* d_ws, size_t ws_size,
                              hipStream_t stream) {
    const float* states  = (const float*)d_in[0];
    const float* weights = (const float*)d_in[1];
    const float* plastic = (const float*)d_in[2];
    float* out        = (float*)d_out;
    float* newStates  = out;
    float* outPlastic = out + NEWSTATES_ELEMS;
    float* wsW = (float*)d_ws;
    float* wsH = wsW + NEDGE * WCNT;

    dim3 tiny((NEDGE * WCNT + 255) / 256);
    prep_kernel<<<tiny, 256, 0, stream>>>(weights, plastic, wsW, wsH);
    conv_agg_kernel<<<dim3(HW, BATCH, NNODE), 256, 0, stream>>>(states, wsW, newStates);
    hebb_kernel<<<dim3(HW / 8, BATCH, NEDGE), 256, 0, stream>>>(states, newStates, wsH);
    finish_kernel<<<tiny, 256, 0, stream>>>(plastic, wsH, outPlastic);
}